// Attention_5282809774349
// MI455X (gfx1250) — compile-verified
//
#include <hip/hip_runtime.h>
#include <hip/hip_bf16.h>
#include <math.h>

// ---------------- problem constants (from reference) ----------------
#define BB      64
#define HDIM    2048
#define NH      16
#define NKV     2
#define HD      128
#define MAXLEN  8192
#define STARTP  8191
#define NREP    8           // NH / NKV
#define NCHUNK  16
#define CHUNK   512         // keys per attention block
#define GITER   8           // 512 / 64 key-groups per block
#define NQKV    2560        // fused q(2048) + k(256) + v(256) output width

typedef __attribute__((ext_vector_type(2))) float v2f;
typedef __attribute__((ext_vector_type(8))) float v8f;

// workspace float offsets
#define WS_RAW   0                                   // B*NQKV       = 163840
#define WS_Q     (WS_RAW  + BB*NQKV)                 // B*NH*HD      = 131072
#define WS_KNEW  (WS_Q    + BB*NH*HD)                // B*NKV*HD     = 16384
#define WS_VNEW  (WS_KNEW + BB*NKV*HD)               // B*NKV*HD     = 16384
#define WS_M     (WS_VNEW + BB*NKV*HD)               // B*NKV*NCHUNK*8
#define WS_L     (WS_M    + BB*NKV*NCHUNK*8)
#define WS_ACC   (WS_L    + BB*NKV*NCHUNK*8)         // B*NKV*NCHUNK*8*128
#define WS_ATTN  (WS_ACC  + BB*NKV*NCHUNK*8*HD)      // B*NH*HD

// ========= kernel 1: fused QKV projection GEMM (WMMA, blocked) ========
// C(64 x 2560) = X(64 x 2048) @ [wq;wk;wv]^T + bias.
// M = 64 = exactly 4 M-tiles of 16 -> each weight row is streamed from
// L2/HBM only 4x (vs 64x for an unblocked kernel): ~84 MB weight traffic.
// Block: 4 waves; wave owns one 16-col N-tile; X tile staged in LDS.
__global__ __launch_bounds__(128)
void gemm_qkv_kernel(const float* __restrict__ x,
                     const float* __restrict__ wq, const float* __restrict__ bq,
                     const float* __restrict__ wk, const float* __restrict__ bk,
                     const float* __restrict__ wv, const float* __restrict__ bv,
                     float* __restrict__ raw)
{
    const int mtile = blockIdx.x / 40;            // 4 M-tiles
    const int nblk  = blockIdx.x % 40;            // 40 N-blocks of 4 tiles
    const int tid   = threadIdx.x;
    const int wave  = tid >> 5;
    const int lane  = tid & 31;
    const int ln    = lane & 15;
    const int half  = lane >> 4;
    const int ntile = nblk * 4 + wave;            // 0..159
    const int col   = ntile * 16 + ln;            // output column for this lane
    const int m0    = mtile * 16;

    // per-lane weight row + bias (tile never straddles matrices: 2048,2304 % 16 == 0)
    const float* wrow;
    float bias;
    if (col < HDIM) {
        wrow = wq + (size_t)col * HDIM;
        bias = bq[col];
    } else if (col < HDIM + NKV * HD) {
        wrow = wk + (size_t)(col - HDIM) * HDIM;
        bias = bk[col - HDIM];
    } else {
        wrow = wv + (size_t)(col - HDIM - NKV * HD) * HDIM;
        bias = bv[col - HDIM - NKV * HD];
    }

    __shared__ float xs[16 * 516];                // padded: conflict-free b64 reads
    v8f c = {};

    for (int seg = 0; seg < HDIM; seg += 512) {
        __syncthreads();                          // previous segment fully consumed
        for (int i = tid; i < (16 * 512) / 4; i += 128) {
            const int e = i * 4;
            const int r = e >> 9, cc = e & 511;
            *(float4*)&xs[r * 516 + cc] =
                *(const float4*)&x[(size_t)(m0 + r) * HDIM + seg + cc];
        }
        __syncthreads();

        const float* bp = wrow + seg + 2 * half;
        #pragma unroll 16
        for (int kk = 0; kk < 128; ++kk) {        // K = 512 per segment, steps of 4
            v2f a  = *(const v2f*)&xs[ln * 516 + 4 * kk + 2 * half];
            v2f bb = *(const v2f*)&bp[4 * kk];
            c = __builtin_amdgcn_wmma_f32_16x16x4_f32(
                    false, a, false, bb, (short)0, c, false, false);
        }
    }

    #pragma unroll
    for (int r = 0; r < 8; ++r)                   // c[r]: rows r / r+8 by lane half
        raw[(size_t)(m0 + r + 8 * half) * NQKV + col] = c[r] + bias;
}

// ============== kernel 2: RoPE + scatter (tiny bandwidth) =============
__global__ __launch_bounds__(64)
void rope_scatter_kernel(const float* __restrict__ raw,
                         const float* __restrict__ cosb, const float* __restrict__ sinb,
                         float* __restrict__ qout, float* __restrict__ kout,
                         float* __restrict__ vout)
{
    const int blk  = blockIdx.x;                  // B * (NH + 2*NKV)
    const int b    = blk / (NH + 2 * NKV);
    const int part = blk % (NH + 2 * NKV);
    const int d    = threadIdx.x;                 // 0..63
    const float* rr = raw + (size_t)b * NQKV;

    const float c0 = cosb[d], c1 = cosb[d + 64];
    const float s0 = sinb[d], s1 = sinb[d + 64];

    if (part < NH) {                              // Q heads: RoPE
        const int h = part;
        const float v0 = rr[h * HD + d];
        const float v1 = rr[h * HD + d + 64];
        float* dst = qout + ((size_t)b * NH + h) * HD;
        dst[d]      = v0 * c0 - v1 * s0;
        dst[d + 64] = v1 * c1 + v0 * s1;
    } else if (part < NH + NKV) {                 // K heads: RoPE
        const int g = part - NH;
        const float v0 = rr[HDIM + g * HD + d];
        const float v1 = rr[HDIM + g * HD + d + 64];
        float* dst = kout + ((size_t)b * NKV + g) * HD;
        dst[d]      = v0 * c0 - v1 * s0;
        dst[d + 64] = v1 * c1 + v0 * s1;
    } else {                                      // V heads: copy
        const int g = part - NH - NKV;
        float* dst = vout + ((size_t)b * NKV + g) * HD;
        dst[d]      = rr[HDIM + NKV * HD + g * HD + d];
        dst[d + 64] = rr[HDIM + NKV * HD + g * HD + d + 64];
    }
}

// ============ kernel 3: flash-decoding attention (WMMA) =============
// grid: B*NKV*NCHUNK blocks, 128 threads (4 waves). Streams the 1.07 GB
// fp32 KV cache (HBM-bound floor ~46us at 23.3 TB/s) with running softmax.
__global__ __launch_bounds__(128)
void attn_chunk_kernel(const float* __restrict__ q,
                       const float* __restrict__ cache_k,
                       const float* __restrict__ cache_v,
                       const float* __restrict__ knew,
                       const float* __restrict__ vnew,
                       float* __restrict__ mOut, float* __restrict__ lOut,
                       float* __restrict__ accOut)
{
    const int blk   = blockIdx.x;                 // (b*NKV+g)*NCHUNK + chunk
    const int chunk = blk % NCHUNK;
    const int bg    = blk / NCHUNK;
    const int g     = bg % NKV;
    const int b     = bg / NKV;

    const int tid  = threadIdx.x;
    const int wave = tid >> 5;
    const int lane = tid & 31;
    const int ln   = lane & 15;
    const int half = lane >> 4;

    __shared__ float qs[16 * 132];                // padded: kills bank conflicts
    __shared__ float stile[16 * 66];              // scores / probs, padded
    __shared__ float mrun[16], lrun[16], sf[16];

    // stage Q (rows 8..15 zero-padded) into LDS
    for (int i = tid; i < 16 * 128; i += 128) {
        const int r = i >> 7, c = i & 127;
        float v = 0.0f;
        if (r < 8) v = q[((size_t)b * NH + g * NREP + r) * HD + c];
        qs[r * 132 + c] = v;
    }
    if (tid < 16) { mrun[tid] = -1e30f; lrun[tid] = 0.0f; }
    __syncthreads();

    v8f acc0 = {};
    v8f acc1 = {};
    const int   n0    = wave * 32;                // this wave's HD slice
    const float scale = 0.08838834764831845f;     // 1/sqrt(128)

    const float* knewRow = knew + ((size_t)b * NKV + g) * HD;
    const float* vnewRow = vnew + ((size_t)b * NKV + g) * HD;

    for (int it = 0; it < GITER; ++it) {
        const int t0 = chunk * CHUNK + it * 64;

        // ---- scores: S = Q(16x128) x K_w^T, one 16x16 tile per wave ----
        {
            const int tkey = t0 + wave * 16 + ln;
            const float* krow = (tkey == STARTP)
                ? knewRow
                : cache_k + (((size_t)b * MAXLEN + tkey) * NKV + g) * HD;
            const float* kp = krow + 2 * half;    // this lane's K-slice

            if (it + 1 < GITER) {                 // global_prefetch_b8 next group
                const int tn = t0 + 64 + wave * 16 + ln;
                __builtin_prefetch(cache_k + (((size_t)b * MAXLEN + tn) * NKV + g) * HD, 0, 1);
                __builtin_prefetch(cache_v + (((size_t)b * MAXLEN + tn) * NKV + g) * HD, 0, 1);
            }

            v8f s = {};
            #pragma unroll
            for (int kk = 0; kk < 32; ++kk) {     // K-dim 128 in steps of 4
                v2f a  = *(const v2f*)&qs[ln * 132 + 4 * kk + 2 * half];
                v2f bb = *(const v2f*)&kp[4 * kk];
                s = __builtin_amdgcn_wmma_f32_16x16x4_f32(
                        false, a, false, bb, (short)0, s, false, false);
            }
            #pragma unroll
            for (int r = 0; r < 8; ++r)           // c[r]: rows r (lanes<16), r+8
                stile[(r + 8 * half) * 66 + wave * 16 + ln] = s[r] * scale;
        }
        __syncthreads();

        // ---- running softmax over this 64-key group (1 thread / row) ----
        if (tid < 16) {
            const float mold = mrun[tid];
            float mx = mold;
            for (int c = 0; c < 64; ++c) mx = fmaxf(mx, stile[tid * 66 + c]);
            const float sfv = __expf(mold - mx);
            float lsum = lrun[tid] * sfv;
            for (int c = 0; c < 64; ++c) {
                const float p = __expf(stile[tid * 66 + c] - mx);
                stile[tid * 66 + c] = p;
                lsum += p;
            }
            mrun[tid] = mx; lrun[tid] = lsum; sf[tid] = sfv;
        }
        __syncthreads();

        // ---- rescale accumulators, then acc += P(16x64) x V(64x32) ----
        #pragma unroll
        for (int r = 0; r < 8; ++r) {
            const float f = sf[r + 8 * half];
            acc0[r] *= f;
            acc1[r] *= f;
        }
        #pragma unroll
        for (int kk = 0; kk < 16; ++kk) {
            v2f a = *(const v2f*)&stile[ln * 66 + 4 * kk + 2 * half];
            const int t = t0 + 4 * kk + 2 * half;
            const float* vr0 = (t == STARTP)
                ? vnewRow
                : cache_v + (((size_t)b * MAXLEN + t) * NKV + g) * HD;
            const float* vr1 = (t + 1 == STARTP)
                ? vnewRow
                : cache_v + (((size_t)b * MAXLEN + t + 1) * NKV + g) * HD;
            v2f b0; b0.x = vr0[n0 + ln];      b0.y = vr1[n0 + ln];
            acc0 = __builtin_amdgcn_wmma_f32_16x16x4_f32(
                       false, a, false, b0, (short)0, acc0, false, false);
            v2f b1; b1.x = vr0[n0 + 16 + ln]; b1.y = vr1[n0 + 16 + ln];
            acc1 = __builtin_amdgcn_wmma_f32_16x16x4_f32(
                       false, a, false, b1, (short)0, acc1, false, false);
        }
        __syncthreads();
    }

    // ---- write per-chunk partials (only the 8 real head rows) ----
    const size_t pbase = (size_t)blk * (8 * HD);
    if (half == 0) {
        #pragma unroll
        for (int r = 0; r < 8; ++r) {
            accOut[pbase + r * HD + n0 + ln]      = acc0[r];
            accOut[pbase + r * HD + n0 + 16 + ln] = acc1[r];
        }
    }
    if (tid < 8) {
        mOut[(size_t)blk * 8 + tid] = mrun[tid];
        lOut[(size_t)blk * 8 + tid] = lrun[tid];
    }
}

// ============= kernel 4: LSE-combine the chunk partials =============
__global__ __launch_bounds__(128)
void attn_reduce_kernel(const float* __restrict__ mIn, const float* __restrict__ lIn,
                        const float* __restrict__ accIn, float* __restrict__ attn)
{
    const int blk = blockIdx.x;                   // B*NH
    const int h = blk % NH, b = blk / NH;
    const int g = h / NREP, r = h % NREP;
    const int d = threadIdx.x;                    // 0..127
    const int bg = b * NKV + g;

    float mtot = -1e30f;
    for (int c = 0; c < NCHUNK; ++c)
        mtot = fmaxf(mtot, mIn[(size_t)(bg * NCHUNK + c) * 8 + r]);

    float ltot = 0.0f, o = 0.0f;
    for (int c = 0; c < NCHUNK; ++c) {
        const size_t idx = (size_t)(bg * NCHUNK + c) * 8 + r;
        const float w = __expf(mIn[idx] - mtot);
        ltot += lIn[idx] * w;
        o    += accIn[idx * HD + d] * w;
    }
    attn[((size_t)b * NH + h) * HD + d] = o / ltot;
}

// ========= kernel 5: output projection GEMM (WMMA, blocked) =========
// out(64 x 2048) = attn(64 x 2048) @ wo^T. Same blocking as gemm_qkv:
// wo is streamed only 4x (64 MB) instead of 64x (1 GB).
__global__ __launch_bounds__(128)
void out_proj_kernel(const float* __restrict__ attn, const float* __restrict__ wo,
                     float* __restrict__ out)
{
    const int mtile = blockIdx.x / 32;            // 4 M-tiles
    const int nblk  = blockIdx.x % 32;            // 32 N-blocks of 4 tiles
    const int tid   = threadIdx.x;
    const int wave  = tid >> 5;
    const int lane  = tid & 31;
    const int ln    = lane & 15;
    const int half  = lane >> 4;
    const int ntile = nblk * 4 + wave;            // 0..127
    const int col   = ntile * 16 + ln;
    const int m0    = mtile * 16;

    const float* wrow = wo + (size_t)col * HDIM;

    __shared__ float xs[16 * 516];
    v8f c = {};

    for (int seg = 0; seg < HDIM; seg += 512) {
        __syncthreads();
        for (int i = tid; i < (16 * 512) / 4; i += 128) {
            const int e = i * 4;
            const int r = e >> 9, cc = e & 511;
            *(float4*)&xs[r * 516 + cc] =
                *(const float4*)&attn[(size_t)(m0 + r) * HDIM + seg + cc];
        }
        __syncthreads();

        const float* bp = wrow + seg + 2 * half;
        #pragma unroll 16
        for (int kk = 0; kk < 128; ++kk) {
            v2f a  = *(const v2f*)&xs[ln * 516 + 4 * kk + 2 * half];
            v2f bb = *(const v2f*)&bp[4 * kk];
            c = __builtin_amdgcn_wmma_f32_16x16x4_f32(
                    false, a, false, bb, (short)0, c, false, false);
        }
    }

    #pragma unroll
    for (int r = 0; r < 8; ++r)
        out[(size_t)(m0 + r + 8 * half) * HDIM + col] = c[r];
}

// ============================ launcher ==============================
extern "C" void kernel_launch(void* const* d_in, const int* in_sizes, int n_in,
                              void* d_out, int out_size, void* d_ws, size_t ws_size,
                              hipStream_t stream)
{
    const float* x       = (const float*)d_in[0];
    const float* cosb    = (const float*)d_in[1];
    const float* sinb    = (const float*)d_in[2];
    const float* wq      = (const float*)d_in[3];
    const float* bq      = (const float*)d_in[4];
    const float* wk      = (const float*)d_in[5];
    const float* bk      = (const float*)d_in[6];
    const float* wv      = (const float*)d_in[7];
    const float* bv      = (const float*)d_in[8];
    const float* wo      = (const float*)d_in[9];
    const float* cache_k = (const float*)d_in[10];
    const float* cache_v = (const float*)d_in[11];
    (void)in_sizes; (void)n_in; (void)out_size; (void)ws_size;

    float* ws   = (float*)d_ws;
    float* raw  = ws + WS_RAW;
    float* qbuf = ws + WS_Q;
    float* knew = ws + WS_KNEW;
    float* vnew = ws + WS_VNEW;
    float* mP   = ws + WS_M;
    float* lP   = ws + WS_L;
    float* accP = ws + WS_ACC;
    float* attn = ws + WS_ATTN;
    float* out  = (float*)d_out;

    // 1. fused QKV GEMM (WMMA, batch-blocked: weights streamed 4x not 64x)
    gemm_qkv_kernel<<<4 * 40, 128, 0, stream>>>(
        x, wq, bq, wk, bk, wv, bv, raw);

    // 2. RoPE + scatter to q / k_new / v_new (inputs never mutated)
    rope_scatter_kernel<<<BB * (NH + 2 * NKV), 64, 0, stream>>>(
        raw, cosb, sinb, qbuf, knew, vnew);

    // 3. flash-decoding over KV cache (position 8191 sourced from k/v_new)
    attn_chunk_kernel<<<BB * NKV * NCHUNK, 128, 0, stream>>>(
        qbuf, cache_k, cache_v, knew, vnew, mP, lP, accP);

    // 4. combine chunk partials
    attn_reduce_kernel<<<BB * NH, 128, 0, stream>>>(mP, lP, accP, attn);

    // 5. output projection GEMM (WMMA, batch-blocked)
    out_proj_kernel<<<4 * 32, 128, 0, stream>>>(attn, wo, out);
}